// APPNP_3951369912454
// MI455X (gfx1250) — compile-verified
//
#include <hip/hip_runtime.h>

typedef __attribute__((ext_vector_type(2))) float v2f;
typedef __attribute__((ext_vector_type(8))) float v8f;

#define N_NODES   100000
#define N_EDGES   3200000
#define IN_FEATS  512
#define HIDDENS   256
#define N_CLASSES 64
#define K_ITERS   10
#define ALPHA     0.1f

// ---------------------------------------------------------------------------
// Degree accumulation: one thread per edge, float atomics (global_atomic_add_f32)
// ---------------------------------------------------------------------------
__global__ void appnp_degree_kernel(const long long* __restrict__ src,
                                    const long long* __restrict__ dst,
                                    float* __restrict__ outdeg,
                                    float* __restrict__ indeg)
{
    int e = blockIdx.x * blockDim.x + threadIdx.x;
    if (e >= N_EDGES) return;
    atomicAdd(&outdeg[src[e]], 1.0f);
    atomicAdd(&indeg[dst[e]], 1.0f);
}

// coeff[e] = rsqrt(max(outdeg[src],1)) * rsqrt(max(indeg[dst],1))
__global__ void appnp_coeff_kernel(const long long* __restrict__ src,
                                   const long long* __restrict__ dst,
                                   const float* __restrict__ outdeg,
                                   const float* __restrict__ indeg,
                                   float* __restrict__ coeff)
{
    int e = blockIdx.x * blockDim.x + threadIdx.x;
    if (e >= N_EDGES) return;
    float ns = rsqrtf(fmaxf(outdeg[src[e]], 1.0f));
    float nd = rsqrtf(fmaxf(indeg[dst[e]], 1.0f));
    coeff[e] = ns * nd;
}

// ---------------------------------------------------------------------------
// FP32 WMMA GEMM with LDS-staged B panel:
//   Out[M,N] = act(A[M,K] @ B[K,N] + bias[N])
//
// Block = 256 threads (8 waves). Block covers 128 rows x 64 cols.
// B is staged per K-chunk (KC=128) into LDS *transposed* (Bs[col][k], stride
// 132 floats) so each lane's K-pair is contiguous -> one ds_load_b64 per
// B fragment. Stride 132 => inner-loop read banks (4*col + k) are conflict-
// free across the 32 lanes.
//
// Fragments (per ISA layouts):
//   A(16x4):  lane l -> row (l&15), K-pair 2*(l>>4)      => global b64 load
//   B(4x16):  lane l -> col (l&15), K-pair 2*(l>>4)      => ds_load_b64
//   C/D(16x16): 8 VGPRs; lane l, vgpr v -> row v+8*(l>>4), col (l&15)
//
// All waves stay alive for __syncthreads(); WMMA work is under a wave-uniform
// rowValid guard so EXEC is all-1s inside WMMA-executing waves.
// M % 16 == 0, N % 64 == 0, K % KC == 0 (holds for 100000/512/256/64).
// ---------------------------------------------------------------------------
#define GEMM_WAVES 8
#define GEMM_KC    128
#define B_STRIDE   132   // 64 x 132 floats = 33,792 B LDS

__global__ void appnp_gemm_wmma_f32(const float* __restrict__ A,
                                    const float* __restrict__ B,
                                    const float* __restrict__ bias,
                                    float* __restrict__ Out,
                                    int M, int K, int N, int do_relu)
{
    __shared__ float Bs[64 * B_STRIDE];

    const int wave = threadIdx.x >> 5;
    const int lane = threadIdx.x & 31;
    const int rowTile = blockIdx.x * GEMM_WAVES + wave;
    const bool rowValid = (rowTile * 16 < M);      // uniform per wave

    const int colBase = blockIdx.y * 64;
    const int m  = lane & 15;
    const int kg = lane >> 4;                      // 0 or 1 (selects K-pair)

    const float* __restrict__ arow =
        A + (size_t)((rowValid ? rowTile : 0) * 16 + m) * (size_t)K + 2 * kg;

    v8f acc0 = {}, acc1 = {}, acc2 = {}, acc3 = {};

    for (int k0 = 0; k0 < K; k0 += GEMM_KC) {
        __syncthreads();
        // Stage B[k0:k0+KC, colBase:colBase+64] into LDS, transposed.
        // float4 along columns: coalesced 256B per 16 lanes.
        for (int idx = threadIdx.x; idx < 16 * GEMM_KC; idx += 256) {
            const int cq = (idx & 15) * 4;         // column quad 0,4,...,60
            const int kk = idx >> 4;               // 0..KC-1
            const float4 v =
                *(const float4*)(B + (size_t)(k0 + kk) * (size_t)N + colBase + cq);
            Bs[(cq + 0) * B_STRIDE + kk] = v.x;
            Bs[(cq + 1) * B_STRIDE + kk] = v.y;
            Bs[(cq + 2) * B_STRIDE + kk] = v.z;
            Bs[(cq + 3) * B_STRIDE + kk] = v.w;
        }
        __syncthreads();

        if (rowValid) {
            const float* bsbase = &Bs[m * B_STRIDE + 2 * kg];
#pragma unroll 4
            for (int kk = 0; kk < GEMM_KC; kk += 4) {
                v2f a = *(const v2f*)(arow + k0 + kk);               // global b64
                v2f b0 = *(const v2f*)(bsbase + kk);                 // ds_load_b64
                v2f b1 = *(const v2f*)(bsbase + 16 * B_STRIDE + kk);
                v2f b2 = *(const v2f*)(bsbase + 32 * B_STRIDE + kk);
                v2f b3 = *(const v2f*)(bsbase + 48 * B_STRIDE + kk);
                acc0 = __builtin_amdgcn_wmma_f32_16x16x4_f32(false, a, false, b0, (short)0, acc0, false, false);
                acc1 = __builtin_amdgcn_wmma_f32_16x16x4_f32(false, a, false, b1, (short)0, acc1, false, false);
                acc2 = __builtin_amdgcn_wmma_f32_16x16x4_f32(false, a, false, b2, (short)0, acc2, false, false);
                acc3 = __builtin_amdgcn_wmma_f32_16x16x4_f32(false, a, false, b3, (short)0, acc3, false, false);
            }
        }
    }

    if (!rowValid) return;

    // Epilogue: bias + optional relu. Element (v, lane) -> row v+8*kg, col t*16+m.
    const size_t rbase = (size_t)(rowTile * 16 + 8 * kg) * (size_t)N;
#define APPNP_STORE_TILE(ACC, T)                                                   \
    {                                                                              \
        const int col = colBase + (T) * 16 + m;                                    \
        const float bv = bias[col];                                                \
        _Pragma("unroll")                                                          \
        for (int v = 0; v < 8; ++v) {                                              \
            float r = ACC[v] + bv;                                                 \
            if (do_relu) r = fmaxf(r, 0.0f);                                       \
            Out[rbase + (size_t)v * (size_t)N + col] = r;                          \
        }                                                                          \
    }
    APPNP_STORE_TILE(acc0, 0)
    APPNP_STORE_TILE(acc1, 1)
    APPNP_STORE_TILE(acc2, 2)
    APPNP_STORE_TILE(acc3, 3)
#undef APPNP_STORE_TILE
}

// ---------------------------------------------------------------------------
// Propagation scatter: 16 threads per edge, each moves a float4 feature chunk.
// Consecutive threads cover consecutive chunks of the same h[src] row => one
// fully coalesced 256B line per edge; atomic adds hit L2-resident agg.
// ---------------------------------------------------------------------------
__global__ void appnp_scatter_kernel(const long long* __restrict__ src,
                                     const long long* __restrict__ dst,
                                     const float* __restrict__ coeff,
                                     const float* __restrict__ hprev,
                                     float* __restrict__ agg)
{
    long long tid = (long long)blockIdx.x * blockDim.x + threadIdx.x;
    long long e = tid >> 4;
    int g = (int)(tid & 15);
    if (e >= (long long)N_EDGES) return;

    long long s = src[e];
    long long d = dst[e];
    float c = coeff[e];

    const float4 v = *(const float4*)(hprev + s * N_CLASSES + g * 4);
    float* o = agg + d * N_CLASSES + g * 4;
    atomicAdd(o + 0, v.x * c);
    atomicAdd(o + 1, v.y * c);
    atomicAdd(o + 2, v.z * c);
    atomicAdd(o + 3, v.w * c);
}

// h_next = (1 - alpha) * agg + alpha * h0
__global__ void appnp_blend_kernel(const float* __restrict__ agg,
                                   const float* __restrict__ h0,
                                   float* __restrict__ hnext)
{
    int i = blockIdx.x * blockDim.x + threadIdx.x;
    if (i >= N_NODES * N_CLASSES) return;
    hnext[i] = (1.0f - ALPHA) * agg[i] + ALPHA * h0[i];
}

// ---------------------------------------------------------------------------
// Host-side orchestration (graph-capture safe: only kernel launches +
// hipMemsetAsync on `stream`).
// ---------------------------------------------------------------------------
extern "C" void kernel_launch(void* const* d_in, const int* in_sizes, int n_in,
                              void* d_out, int out_size, void* d_ws, size_t ws_size,
                              hipStream_t stream)
{
    const float*     X   = (const float*)d_in[0];
    const long long* ei  = (const long long*)d_in[1];   // int64 per reference
    const float*     W1  = (const float*)d_in[2];
    const float*     b1  = (const float*)d_in[3];
    const float*     W2  = (const float*)d_in[4];
    const float*     b2  = (const float*)d_in[5];
    float*           out = (float*)d_out;

    const long long* src = ei;
    const long long* dst = ei + N_EDGES;

    // Workspace layout (bytes), total ~141.6 MB:
    //   [0,       400000)  out_deg
    //   [401408,  801408)  in_deg
    //   [802816, +12.8MB)  coeff
    //   [13602816, +102.4MB) H1  -- reused after GEMM2 as {hA, hB, agg}
    //   [116002816, +25.6MB) h0
    char* ws = (char*)d_ws;
    float* outdeg = (float*)(ws + 0);
    float* indeg  = (float*)(ws + 401408);
    float* coeff  = (float*)(ws + 802816);
    float* H1     = (float*)(ws + 13602816);
    float* hA     = H1;                                       // alias (post-GEMM2)
    float* hB     = (float*)(ws + 13602816 + 25600000);
    float* agg    = (float*)(ws + 13602816 + 51200000);
    float* h0     = (float*)(ws + 116002816);

    const int HB = N_NODES * N_CLASSES;                       // 6,400,000 floats

    // --- degrees + edge coefficients -------------------------------------
    hipMemsetAsync(ws, 0, 802816, stream);                    // zero both degree arrays
    {
        int blocks = (N_EDGES + 255) / 256;
        appnp_degree_kernel<<<blocks, 256, 0, stream>>>(src, dst, outdeg, indeg);
        appnp_coeff_kernel<<<blocks, 256, 0, stream>>>(src, dst, outdeg, indeg, coeff);
    }

    // --- MLP: H1 = relu(X@W1 + b1); h0 = H1@W2 + b2 ----------------------
    {
        int rowTiles = N_NODES / 16;                          // 6250
        int gx = (rowTiles + GEMM_WAVES - 1) / GEMM_WAVES;    // 782
        appnp_gemm_wmma_f32<<<dim3(gx, HIDDENS / 64), 256, 0, stream>>>(
            X, W1, b1, H1, N_NODES, IN_FEATS, HIDDENS, 1);
        appnp_gemm_wmma_f32<<<dim3(gx, N_CLASSES / 64), 256, 0, stream>>>(
            H1, W2, b2, h0, N_NODES, HIDDENS, N_CLASSES, 0);
    }

    // --- K iterations of propagate + teleport blend ----------------------
    {
        long long sthreads = (long long)N_EDGES * 16;
        int sblocks = (int)((sthreads + 255) / 256);
        int bblocks = (HB + 255) / 256;

        const float* cur = h0;                                // iteration 0 reads h0
        for (int k = 0; k < K_ITERS; ++k) {
            hipMemsetAsync(agg, 0, (size_t)HB * sizeof(float), stream);
            appnp_scatter_kernel<<<sblocks, 256, 0, stream>>>(src, dst, coeff, cur, agg);
            float* nxt = (k == K_ITERS - 1) ? out : ((k & 1) ? hB : hA);
            appnp_blend_kernel<<<bblocks, 256, 0, stream>>>(agg, h0, nxt);
            cur = nxt;
        }
    }
}